// SQAE_78108275245157
// MI455X (gfx1250) — compile-verified
//
#include <hip/hip_runtime.h>

// ---------------------------------------------------------------------------
// VQ autoencoder forward on gfx1250 (CDNA5), fp32 WMMA path.
//   z = x @ enc_w^T + enc_b                      [8192 x 256]
//   idx = argmax_k (z . emb_k) / ||emb_k||       (== argmin cosine distance)
//   z_q = emb[idx]                               [8192 x 256]
//   x_recon = z_q @ dec_w^T + dec_b              [8192 x 1024]
// Outputs concatenated: x_recon | z | z_q | idx(int32)
// ---------------------------------------------------------------------------

typedef __attribute__((ext_vector_type(2))) float v2f;
typedef __attribute__((ext_vector_type(8))) float v8f;

#define GLOBAL_AS __attribute__((address_space(1)))
typedef const GLOBAL_AS float* gcfp;   // global const float*
typedef GLOBAL_AS float*       gfp;    // global float*
typedef const GLOBAL_AS v2f*   gcv2fp; // global const float2*

#define BATCH   8192
#define IN_DIM  1024
#define LAT     256
#define NEMB    8192

__device__ __forceinline__ unsigned int orderable_f32(float x) {
    unsigned int u = __float_as_uint(x);
    return (u & 0x80000000u) ? ~u : (u | 0x80000000u);
}

// ---------------------------------------------------------------------------
// Register-blocked, software-pipelined fp32 WMMA GEMM core.
// One wave accumulates a (16*MT) x (16*NT) tile of  A(MxK) * W(NxK)^T.
// A frags reused NT times, B frags reused MT times. K-loop steps 16 with
// 4 immediate-offset sub-steps; fragments are double-buffered so ~MT+NT
// global_load_b64 are always in flight underneath each 16-WMMA block.
// V_WMMA_F32_16X16X4_F32 fragment layout (lm=lane&15, h=lane>>4):
//   A: float2 = A[m][k0+2h .. k0+2h+1]     B: float2 = W[n][k0+2h .. k0+2h+1]
//   C/D vgpr r: row +r+8h, col +lm
// ---------------------------------------------------------------------------
template <int MT, int NT, int K>
__device__ __forceinline__ void gemm_core(
    const float* __restrict__ A, const float* __restrict__ W,
    int m0, int n0, int lm, int h, v8f (&acc)[MT][NT])
{
    gcfp ap[MT];
    gcfp bp[NT];
#pragma unroll
    for (int i = 0; i < MT; ++i)
        ap[i] = (gcfp)(A + (size_t)(m0 + i * 16 + lm) * K + 2 * h);
#pragma unroll
    for (int t = 0; t < NT; ++t)
        bp[t] = (gcfp)(W + (size_t)(n0 + t * 16 + lm) * K + 2 * h);

    v2f a0[MT], b0[NT], a1[MT], b1[NT];

#define LOADG(av, bv, off)                                                   \
    {                                                                        \
        _Pragma("unroll")                                                    \
        for (int i = 0; i < MT; ++i) av[i] = *(gcv2fp)(ap[i] + (off));       \
        _Pragma("unroll")                                                    \
        for (int t = 0; t < NT; ++t) bv[t] = *(gcv2fp)(bp[t] + (off));       \
    }
#define MMAG(av, bv)                                                         \
    {                                                                        \
        _Pragma("unroll")                                                    \
        for (int i = 0; i < MT; ++i) {                                       \
            _Pragma("unroll")                                                \
            for (int t = 0; t < NT; ++t)                                     \
                acc[i][t] = __builtin_amdgcn_wmma_f32_16x16x4_f32(           \
                    false, av[i], false, bv[t], (short)0, acc[i][t],         \
                    false, false);                                           \
        }                                                                    \
    }

    LOADG(a0, b0, 0);
    for (int kk = 0; kk < K / 16 - 1; ++kk) {
        LOADG(a1, b1, 4);  MMAG(a0, b0);
        LOADG(a0, b0, 8);  MMAG(a1, b1);
        LOADG(a1, b1, 12); MMAG(a0, b0);
#pragma unroll
        for (int i = 0; i < MT; ++i) ap[i] += 16;
#pragma unroll
        for (int t = 0; t < NT; ++t) bp[t] += 16;
        LOADG(a0, b0, 0);  MMAG(a1, b1);
    }
    // final 16-K group (no further prefetch past the buffer end)
    LOADG(a1, b1, 4);  MMAG(a0, b0);
    LOADG(a0, b0, 8);  MMAG(a1, b1);
    LOADG(a1, b1, 12); MMAG(a0, b0);
    MMAG(a1, b1);

#undef LOADG
#undef MMAG
}

// ---------------------------------------------------------------------------
// GEMM + bias kernel:  out[m,n] = sum_k A[m,k]*W[n,k] + bias[n]
// ---------------------------------------------------------------------------
template <int MT, int NT, int K>
__global__ __launch_bounds__(256) void wmma_gemm_bias(
    const float* __restrict__ A, const float* __restrict__ W,
    const float* __restrict__ bias, float* __restrict__ out,
    int N, int Mtiles)
{
    const int wave = blockIdx.x * 8 + (threadIdx.x >> 5);
    const int lane = threadIdx.x & 31;
    const int h    = lane >> 4;
    const int lm   = lane & 15;
    const int m0 = (wave % Mtiles) * (16 * MT);
    const int n0 = (wave / Mtiles) * (16 * NT);

    v8f acc[MT][NT] = {};
    gemm_core<MT, NT, K>(A, W, m0, n0, lm, h, acc);

    gfp outg = (gfp)out;
    const gcfp biasg = (gcfp)bias;
#pragma unroll
    for (int t = 0; t < NT; ++t) {
        const int col = n0 + t * 16 + lm;
        const float bv = biasg[col];
#pragma unroll
        for (int i = 0; i < MT; ++i)
#pragma unroll
            for (int r = 0; r < 8; ++r)
                outg[(size_t)(m0 + i * 16 + r + 8 * h) * N + col] =
                    acc[i][t][r] + bv;
    }
}

// ---------------------------------------------------------------------------
// Scores + fused argmax:  s[m,n] = (z[m,:] . emb[n,:]) * einv[n]
// Wave computes 32 rows x 128 cols (MT=2, NT=8), reduces per-row best
// (max score, ties -> lowest index) and publishes one packed u64 atomicMax
// per row:  key = orderable(score)<<32 | (NEMB-1-col).
// ---------------------------------------------------------------------------
__global__ __launch_bounds__(256) void scores_argmax(
    const float* __restrict__ z, const float* __restrict__ emb,
    const float* __restrict__ einv, unsigned long long* __restrict__ keys)
{
    constexpr int MT = 2, NT = 8;
    const int wave = blockIdx.x * 8 + (threadIdx.x >> 5);
    const int lane = threadIdx.x & 31;
    const int h    = lane >> 4;
    const int lm   = lane & 15;
    const int Mtiles = BATCH / (16 * MT);        // 256
    const int m0 = (wave % Mtiles) * (16 * MT);
    const int n0 = (wave / Mtiles) * (16 * NT);  // 64 strips of 128 cols

    v8f acc[MT][NT] = {};
    gemm_core<MT, NT, LAT>(z, emb, m0, n0, lm, h, acc);

    // per-lane column scales (depend only on lm)
    const gcfp einvg = (gcfp)einv;
    float ein[NT];
#pragma unroll
    for (int t = 0; t < NT; ++t) ein[t] = einvg[n0 + t * 16 + lm];

#pragma unroll
    for (int i = 0; i < MT; ++i) {
#pragma unroll
        for (int r = 0; r < 8; ++r) {
            float best = -3.402823466e38f;
            int   bcol = 0;
#pragma unroll
            for (int t = 0; t < NT; ++t) {
                float v = acc[i][t][r] * ein[t];
                int   c = n0 + t * 16 + lm;
                if (v > best || (v == best && c < bcol)) { best = v; bcol = c; }
            }
            // reduce across the 16 lanes of this half (rows differ per half)
#pragma unroll
            for (int m = 8; m >= 1; m >>= 1) {
                float ov = __shfl_xor(best, m, 32);
                int   oc = __shfl_xor(bcol, m, 32);
                if (ov > best || (ov == best && oc < bcol)) { best = ov; bcol = oc; }
            }
            if (lm == 0) {
                unsigned long long key =
                    ((unsigned long long)orderable_f32(best) << 32) |
                    (unsigned int)(NEMB - 1 - bcol);
                atomicMax(&keys[m0 + i * 16 + r + 8 * h], key);
            }
        }
    }
}

// ---------------------------------------------------------------------------
// Inverse L2 norms of codebook rows: einv[k] = rsqrt(sum(emb[k,:]^2)).
// One wave per code row.
// ---------------------------------------------------------------------------
__global__ __launch_bounds__(256) void emb_inv_norms(
    const float* __restrict__ emb, float* __restrict__ einv)
{
    const int wave = blockIdx.x * 8 + (threadIdx.x >> 5);
    const int lane = threadIdx.x & 31;
    const gcfp row = (gcfp)(emb + (size_t)wave * LAT);
    float s = 0.0f;
#pragma unroll
    for (int i = 0; i < LAT / 32; ++i) {
        float v = row[lane + 32 * i];
        s += v * v;
    }
#pragma unroll
    for (int m = 16; m >= 1; m >>= 1)
        s += __shfl_xor(s, m, 32);
    if (lane == 0) ((gfp)einv)[wave] = rsqrtf(s);
}

__global__ __launch_bounds__(256) void init_keys(unsigned long long* keys)
{
    int i = blockIdx.x * 256 + threadIdx.x;
    if (i < BATCH) keys[i] = 0ull;
}

// ---------------------------------------------------------------------------
// Decode packed keys -> idx, gather z_q = emb[idx]. One thread per element.
// ---------------------------------------------------------------------------
__global__ __launch_bounds__(256) void decode_gather(
    const unsigned long long* __restrict__ keys,
    const float* __restrict__ emb,
    float* __restrict__ zq, int* __restrict__ idx_out)
{
    const int gid = blockIdx.x * 256 + threadIdx.x;   // BATCH*LAT threads
    const int row = gid >> 8;                         // / LAT
    const int e   = gid & (LAT - 1);
    const unsigned long long k = keys[row];
    const int idx = NEMB - 1 - (int)(unsigned int)(k & 0xFFFFFFFFu);
    ((gfp)zq)[(size_t)row * LAT + e] = ((gcfp)emb)[(size_t)idx * LAT + e];
    if (e == 0) idx_out[row] = idx;
}

// ---------------------------------------------------------------------------
extern "C" void kernel_launch(void* const* d_in, const int* in_sizes, int n_in,
                              void* d_out, int out_size, void* d_ws, size_t ws_size,
                              hipStream_t stream)
{
    const float* x     = (const float*)d_in[0];
    const float* enc_w = (const float*)d_in[1];
    const float* enc_b = (const float*)d_in[2];
    const float* emb   = (const float*)d_in[3];
    const float* dec_w = (const float*)d_in[4];
    const float* dec_b = (const float*)d_in[5];

    float* out_base = (float*)d_out;
    float* x_recon  = out_base;                                   // 8192*1024
    float* z_out    = out_base + (size_t)BATCH * IN_DIM;          // 8192*256
    float* zq_out   = z_out    + (size_t)BATCH * LAT;             // 8192*256
    int*   idx_out  = (int*)(zq_out + (size_t)BATCH * LAT);       // 8192

    unsigned long long* keys = (unsigned long long*)d_ws;         // 64 KB
    float* einv = (float*)((char*)d_ws + BATCH * sizeof(unsigned long long));

    // 1. clear argmax keys
    init_keys<<<BATCH / 256, 256, 0, stream>>>(keys);

    // 2. codebook inverse norms: 8192 waves, 8 waves/block
    emb_inv_norms<<<NEMB / 8, 256, 0, stream>>>(emb, einv);

    // 3. encoder GEMM: M=8192 (256 tiles of 32), N=256 (4 strips of 64), K=1024
    //    1024 waves -> 128 blocks
    wmma_gemm_bias<2, 4, IN_DIM><<<128, 256, 0, stream>>>(
        x, enc_w, enc_b, z_out, LAT, BATCH / 32);

    // 4. scores + fused argmax: 256 Mtiles * 64 strips = 16384 waves -> 2048 blocks
    scores_argmax<<<2048, 256, 0, stream>>>(z_out, emb, einv, keys);

    // 5. decode idx + gather z_q
    decode_gather<<<(BATCH * LAT) / 256, 256, 0, stream>>>(
        keys, emb, zq_out, idx_out);

    // 6. decoder GEMM: M=8192 (256 tiles of 32), N=1024 (16 strips of 64), K=256
    //    4096 waves -> 512 blocks
    wmma_gemm_bias<2, 4, LAT><<<512, 256, 0, stream>>>(
        zq_out, dec_w, dec_b, x_recon, IN_DIM, BATCH / 32);
}